// Net_63917703299746
// MI455X (gfx1250) — compile-verified
//
#include <hip/hip_runtime.h>

#define NN 50000
#define EE 800000
#define FF 64
#define PP 32
#define GG 64
#define MAXD 10

typedef __attribute__((ext_vector_type(2))) float v2f;
typedef __attribute__((ext_vector_type(8))) float v8f;

__global__ void fill_f32(float* __restrict__ p, float v, int n) {
  int i = blockIdx.x * blockDim.x + threadIdx.x;
  if (i < n) p[i] = v;
}

// Degree with self-loop baked in (cntdeg pre-filled to 1.0), plus raw edge degree for MFConv.
__global__ void deg_kernel(const int* __restrict__ dst, float* __restrict__ cntdeg,
                           int* __restrict__ degm, int E) {
  int e = blockIdx.x * blockDim.x + threadIdx.x;
  if (e < E) {
    int d = dst[e];
    unsafeAtomicAdd(&cntdeg[d], 1.0f);
    atomicAdd(&degm[d], 1);
  }
}

__global__ void dinv_kernel(const float* __restrict__ cntdeg, float* __restrict__ dinv,
                            const int* __restrict__ batch, float* __restrict__ cnt, int N) {
  int n = blockIdx.x * blockDim.x + threadIdx.x;
  if (n < N) {
    dinv[n] = rsqrtf(cntdeg[n]);                 // cntdeg >= 1 (self loop)
    unsafeAtomicAdd(&cnt[batch[n]], 1.0f);       // nodes per graph
  }
}

// C[N x Fo] = A[N x 64] * W[64 x Fo], fp32 WMMA 16x16x4.
// One wave per 16-column strip; B tile (all K=64) preloaded into 32 VGPRs once,
// then reused across RT row tiles. Fo is compile-time so all W loads / C stores
// are immediate-offset. Grid covers exactly N/ (16*RT) blocks (no tail, EXEC all-ones).
template <int Fo, int RT>
__global__ void gemm_wmma_f32(const float* __restrict__ A, const float* __restrict__ W,
                              float* __restrict__ C) {
  int wave = threadIdx.x >> 5;
  int lane = threadIdx.x & 31;
  int half = lane >> 4;        // 0: K={0,1} of each step, 1: K={2,3}
  int l16  = lane & 15;
  int col0 = wave << 4;

  // Preload full B strip: 16 k-steps x v2f = 32 VGPRs, immediate offsets.
  const float* wcol = W + col0 + l16;
  v2f bfrag[16];
#pragma unroll
  for (int kb = 0; kb < 16; ++kb) {
    int ka = 4 * kb + 2 * half;
    v2f b; b.x = wcol[ka * Fo]; b.y = wcol[(ka + 1) * Fo];
    bfrag[kb] = b;
  }

#pragma unroll
  for (int r = 0; r < RT; ++r) {
    int tile = blockIdx.x * RT + r;
    const float* arow = A + (tile * 16 + l16) * FF + 2 * half;
    v8f acc = {};
#pragma unroll
    for (int kb = 0; kb < 16; ++kb) {
      float2 av = *(const float2*)(arow + 4 * kb);   // 8B aligned immediate offset
      v2f a; a.x = av.x; a.y = av.y;
      acc = __builtin_amdgcn_wmma_f32_16x16x4_f32(
          /*neg_a=*/false, a, /*neg_b=*/false, bfrag[kb],
          /*c_mod=*/(short)0, acc, /*reuse_a=*/false, /*reuse_b=*/false);
    }
    // D layout: VGPR rr holds row m = 8*half + rr, col = l16
    float* cc = C + (tile * 16 + 8 * half) * Fo + col0 + l16;
#pragma unroll
    for (int rr = 0; rr < 8; ++rr) cc[rr * Fo] = acc[rr];
  }
}

// out[n][f] = dinv[n]^2 * hw[n][f] + b[f]   (self-loop term + bias, pre-scatter)
template <int Fo>
__global__ void init_self(const float* __restrict__ hw, const float* __restrict__ dinv,
                          const float* __restrict__ b, float* __restrict__ out) {
  constexpr int TPE = Fo >> 2;
  int tid = blockIdx.x * blockDim.x + threadIdx.x;
  if (tid >= NN * TPE) return;
  int node = tid / TPE;                // power-of-two -> shift
  int fc = (tid - node * TPE) << 2;
  float s = dinv[node]; s *= s;
  float4 v = *(const float4*)(hw + node * Fo + fc);
  float4 bb = *(const float4*)(b + fc);
  float4 o;
  o.x = fmaf(s, v.x, bb.x); o.y = fmaf(s, v.y, bb.y);
  o.z = fmaf(s, v.z, bb.z); o.w = fmaf(s, v.w, bb.w);
  *(float4*)(out + node * Fo + fc) = o;
}

// out[dst] += (NORM ? dinv[src]*dinv[dst] : 1) * hin[src]; 4 floats per thread.
template <int Fo, bool NORM>
__global__ void scatter_edges(const int* __restrict__ src, const int* __restrict__ dst,
                              const float* __restrict__ dinv, const float* __restrict__ hin,
                              float* __restrict__ out) {
  constexpr int TPE = Fo >> 2;
  int tid = blockIdx.x * blockDim.x + threadIdx.x;
  if (tid >= EE * TPE) return;
  int e = tid / TPE;                   // power-of-two -> shift
  int fc = (tid - e * TPE) << 2;
  int s = src[e], d = dst[e];
  float nrm = 1.0f;
  if (NORM) nrm = dinv[s] * dinv[d];
  float4 v = *(const float4*)(hin + s * Fo + fc);
  float* o = out + d * Fo + fc;
  unsafeAtomicAdd(o + 0, nrm * v.x);
  unsafeAtomicAdd(o + 1, nrm * v.y);
  unsafeAtomicAdd(o + 2, nrm * v.z);
  unsafeAtomicAdd(o + 3, nrm * v.w);
}

// One wave32 per node: val[n] = <hsum[n], Wl[deg]> + bl[deg] + <h2[n], Wr[deg]>,
// pooled per graph with an atomic.
__global__ void mf_val(const float* __restrict__ h2, const float* __restrict__ hsum,
                       const int* __restrict__ degm, const float* __restrict__ Wl,
                       const float* __restrict__ bl, const float* __restrict__ Wr,
                       const int* __restrict__ batch, float* __restrict__ val_out) {
  int node = blockIdx.x * (blockDim.x >> 5) + (threadIdx.x >> 5);
  if (node >= NN) return;
  int lane = threadIdx.x & 31;
  int deg = degm[node]; deg = deg > MAXD ? MAXD : deg;
  const float* wl = Wl + deg * FF;
  const float* wr = Wr + deg * FF;
  float acc = 0.0f;
#pragma unroll
  for (int f = lane; f < FF; f += 32)
    acc = fmaf(hsum[node * FF + f], wl[f], fmaf(h2[node * FF + f], wr[f], acc));
#pragma unroll
  for (int off = 16; off; off >>= 1) acc += __shfl_down(acc, off, 32);
  if (lane == 0) unsafeAtomicAdd(&val_out[batch[node]], acc + bl[deg]);
}

__global__ void pol_pool(const float* __restrict__ poln, const int* __restrict__ batch,
                         float* __restrict__ polsum) {
  int tid = blockIdx.x * blockDim.x + threadIdx.x;
  if (tid >= NN * PP) return;
  int node = tid >> 5;            // PP == 32
  int p = tid & 31;
  unsafeAtomicAdd(&polsum[batch[node] * PP + p], poln[node * PP + p]);
}

__global__ void finalize_pol(const float* __restrict__ polsum, const float* __restrict__ cnt,
                             float* __restrict__ out) {
  int tid = blockIdx.x * blockDim.x + threadIdx.x;
  if (tid >= GG * PP) return;
  int g = tid >> 5;
  out[tid] = polsum[tid] / fmaxf(cnt[g], 1.0f);
}

extern "C" void kernel_launch(void* const* d_in, const int* in_sizes, int n_in,
                              void* d_out, int out_size, void* d_ws, size_t ws_size,
                              hipStream_t stream) {
  const float* x     = (const float*)d_in[0];
  const int*   ei    = (const int*)d_in[1];
  const int*   src   = ei;              // edge_index[0]
  const int*   dst   = ei + EE;         // edge_index[1]
  const int*   batch = (const int*)d_in[2];
  const float* W_in  = (const float*)d_in[3];
  const float* b_in  = (const float*)d_in[4];
  const float* W1    = (const float*)d_in[5];
  const float* b1    = (const float*)d_in[6];
  const float* Wl    = (const float*)d_in[7];
  const float* bl    = (const float*)d_in[8];
  const float* Wr    = (const float*)d_in[9];
  const float* W_pol = (const float*)d_in[10];
  const float* b_pol = (const float*)d_in[11];
  float* out = (float*)d_out;           // [0,2048): pol, [2048,2112): val

  float* ws     = (float*)d_ws;
  float* cntdeg = ws;                   // N
  float* dinv   = ws + NN;              // N
  int*   degm   = (int*)(ws + 2 * NN);  // N
  float* buf0   = ws + 3 * NN;          // N*F : xw, reused as h1w
  float* buf1   = buf0 + NN * FF;       // N*F : h1, reused as hsum
  float* buf2   = buf1 + NN * FF;       // N*F : h2
  float* polw   = buf2 + NN * FF;       // N*P
  float* poln   = polw + NN * PP;       // N*P
  float* polsum = poln + NN * PP;       // G*P
  float* cnt    = polsum + GG * PP;     // G

  auto nb = [](int n, int t) { return (n + t - 1) / t; };

  // --- init accumulators ---
  fill_f32<<<nb(NN, 256), 256, 0, stream>>>(cntdeg, 1.0f, NN);          // self-loop degree
  fill_f32<<<nb(NN, 256), 256, 0, stream>>>((float*)degm, 0.0f, NN);    // int 0 bit pattern
  fill_f32<<<nb(GG * PP + GG, 256), 256, 0, stream>>>(polsum, 0.0f, GG * PP + GG);
  fill_f32<<<nb(GG * PP + GG, 256), 256, 0, stream>>>(out, 0.0f, GG * PP + GG);

  // --- degrees / norms ---
  deg_kernel<<<nb(EE, 256), 256, 0, stream>>>(dst, cntdeg, degm, EE);
  dinv_kernel<<<nb(NN, 256), 256, 0, stream>>>(cntdeg, dinv, batch, cnt, NN);

  // 3125 row tiles = 625 blocks x 5 tiles/wave (exact)
  const int GEMM_GRID = NN / (16 * 5);  // 625

  // --- GCN layer 1: h1 = norm-scatter(x @ W_in) + self + bias ---
  gemm_wmma_f32<FF, 5><<<GEMM_GRID, 32 * (FF / 16), 0, stream>>>(x, W_in, buf0);
  init_self<FF><<<nb(NN * FF / 4, 256), 256, 0, stream>>>(buf0, dinv, b_in, buf1);
  scatter_edges<FF, true><<<nb(EE * FF / 4, 256), 256, 0, stream>>>(src, dst, dinv, buf0, buf1);

  // --- GCN layer 2: h2 = norm-scatter(h1 @ W1) + self + bias ---
  gemm_wmma_f32<FF, 5><<<GEMM_GRID, 32 * (FF / 16), 0, stream>>>(buf1, W1, buf0);
  init_self<FF><<<nb(NN * FF / 4, 256), 256, 0, stream>>>(buf0, dinv, b1, buf2);
  scatter_edges<FF, true><<<nb(EE * FF / 4, 256), 256, 0, stream>>>(src, dst, dinv, buf0, buf2);

  // --- MFConv neighbor sum: hsum = scatter(h2) (buf1 reused) ---
  fill_f32<<<nb(NN * FF, 256), 256, 0, stream>>>(buf1, 0.0f, NN * FF);
  scatter_edges<FF, false><<<nb(EE * FF / 4, 256), 256, 0, stream>>>(src, dst, nullptr, buf2, buf1);

  // --- policy GCN: poln = norm-scatter(h2 @ W_pol) + self + bias ---
  gemm_wmma_f32<PP, 5><<<GEMM_GRID, 32 * (PP / 16), 0, stream>>>(buf2, W_pol, polw);
  init_self<PP><<<nb(NN * PP / 4, 256), 256, 0, stream>>>(polw, dinv, b_pol, poln);
  scatter_edges<PP, true><<<nb(EE * PP / 4, 256), 256, 0, stream>>>(src, dst, dinv, polw, poln);

  // --- value head + pooling ---
  mf_val<<<nb(NN, 8), 256, 0, stream>>>(buf2, buf1, degm, Wl, bl, Wr, batch, out + GG * PP);
  pol_pool<<<nb(NN * PP, 256), 256, 0, stream>>>(poln, batch, polsum);
  finalize_pol<<<nb(GG * PP, 256), 256, 0, stream>>>(polsum, cnt, out);
}